// botDetect_44736379355437
// MI455X (gfx1250) — compile-verified
//
#include <hip/hip_runtime.h>
#include <hip/hip_bf16.h>

typedef __attribute__((ext_vector_type(2))) float v2f;
typedef __attribute__((ext_vector_type(8))) float v8f;

#define NEG_SLOPE 0.2f

// ---------- helpers ----------

__device__ __forceinline__ float warp_sum(float v) {
  #pragma unroll
  for (int o = 16; o > 0; o >>= 1) v += __shfl_xor(v, o, 32);
  return v;
}

// monotonic float -> uint key so integer atomicMax == float max
__device__ __forceinline__ unsigned fkey(float f) {
  unsigned u = __float_as_uint(f);
  return (u & 0x80000000u) ? ~u : (u | 0x80000000u);
}
__device__ __forceinline__ float funkey(unsigned k) {
  unsigned u = (k & 0x80000000u) ? (k & 0x7FFFFFFFu) : ~k;
  return __uint_as_float(u);
}

__device__ __forceinline__ void atomic_add_f32(float* p, float v) {
#if defined(__HIP_DEVICE_COMPILE__)
  unsafeAtomicAdd(p, v);   // lowers to global_atomic_add_f32
#else
  atomicAdd(p, v);
#endif
}

// ---------- init ----------

__global__ void zero_f32(float* __restrict__ p, size_t n) {
  size_t stride = (size_t)gridDim.x * blockDim.x;
  for (size_t i = (size_t)blockIdx.x * blockDim.x + threadIdx.x; i < n; i += stride)
    p[i] = 0.0f;
}

// ---------- layer-1 GEMM: H[N,128] = X[N,256] @ W[256,128], fp32 WMMA ----------
// grid.x = row tiles (16 rows each), block = 256 threads = 8 waves, wave w -> cols [16w,16w+16)
__global__ void gemm1_wmma(const float* __restrict__ X, const float* __restrict__ W,
                           float* __restrict__ H, int N) {
  const int lane = threadIdx.x & 31;
  const int wave = threadIdx.x >> 5;
  const int row0 = blockIdx.x * 16;
  const int col0 = wave * 16;
  const int m  = lane & 15;   // A: row in tile; B/C: col in tile
  const int kh = lane >> 4;   // 0 -> K{0,1}, 1 -> K{2,3}

  int row = row0 + m; if (row >= N) row = N - 1;   // clamp for ragged tail (unused: N%16==0)
  const float* xrow = X + (size_t)row * 256;

  v8f c = {};
#if __has_builtin(__builtin_amdgcn_wmma_f32_16x16x4_f32)
  #pragma unroll 4
  for (int k = 0; k < 256; k += 4) {
    // A 16x4 fp32: VGPR0 = K=(k+2*kh), VGPR1 = K=(k+2*kh+1)  (ISA 7.12.2)
    v2f a = *(const v2f*)(xrow + k + 2 * kh);
    // B 4x16 fp32: same K split across lane halves, N = m
    v2f b;
    const float* wp = W + (size_t)(k + 2 * kh) * 128 + col0 + m;
    b.x = wp[0];
    b.y = wp[128];
    c = __builtin_amdgcn_wmma_f32_16x16x4_f32(false, a, false, b, (short)0, c,
                                              false, false);
  }
#else
  for (int k = 0; k < 256; k += 4) {  // scalar fallback (not taken on gfx1250)
    v2f a = *(const v2f*)(xrow + k + 2 * kh);
    const float* wp = W + (size_t)(k + 2 * kh) * 128 + col0 + m;
    c[0] += a.x * wp[0] + a.y * wp[128];
  }
#endif
  // C 16x16 fp32: VGPR r -> row (r + 8*kh), col m. Uniform fast path: full tile.
  float* hp = H + (size_t)(row0 + 8 * kh) * 128 + col0 + m;
  if (row0 + 16 <= N) {
    #pragma unroll
    for (int r = 0; r < 8; ++r) hp[(size_t)r * 128] = c[r];
  } else {
    #pragma unroll
    for (int r = 0; r < 8; ++r)
      if (row0 + r + 8 * kh < N) hp[(size_t)r * 128] = c[r];
  }
}

// ---------- per-node attention coefficients, F=128 (wave per node) ----------
__global__ void node_e1(const float* __restrict__ H, const float* __restrict__ as,
                        const float* __restrict__ ad, float* __restrict__ es,
                        float* __restrict__ ed, int N) {
  int n = blockIdx.x * 8 + (threadIdx.x >> 5);
  if (n >= N) return;
  int lane = threadIdx.x & 31;
  float4 hv = ((const float4*)(H + (size_t)n * 128))[lane];
  float4 sv = ((const float4*)as)[lane];
  float4 dv = ((const float4*)ad)[lane];
  float s = hv.x * sv.x + hv.y * sv.y + hv.z * sv.z + hv.w * sv.w;
  float d = hv.x * dv.x + hv.y * dv.y + hv.z * dv.z + hv.w * dv.w;
  s = warp_sum(s);
  d = warp_sum(d);
  if (lane == 0) { es[n] = s; ed[n] = d; }
}

// ---------- edge pass 1: leaky-relu logits + segment max ----------
__global__ void edge_logits(const float* __restrict__ es, const float* __restrict__ ed,
                            const int* __restrict__ src, const int* __restrict__ dst,
                            int E, int nT, float* __restrict__ ebuf,
                            unsigned* __restrict__ mkey) {
  int i = blockIdx.x * blockDim.x + threadIdx.x;
  if (i >= nT) return;
  int s, d;
  if (i < E) { s = src[i]; d = dst[i]; } else { s = d = i - E; }
  float e = es[s] + ed[d];
  e = (e > 0.0f) ? e : NEG_SLOPE * e;
  ebuf[i] = e;
  atomicMax(&mkey[d], fkey(e));
}

// ---------- edge pass 2: exp(e - max) + segment sum ----------
__global__ void edge_exp(const float* __restrict__ ein, const unsigned* __restrict__ mkey,
                         const int* __restrict__ dst, int E, int nT,
                         float* __restrict__ eout, float* __restrict__ denom) {
  int i = blockIdx.x * blockDim.x + threadIdx.x;
  if (i >= nT) return;
  int d = (i < E) ? dst[i] : (i - E);
  float ex = expf(ein[i] - funkey(mkey[d]));
  eout[i] = ex;
  atomic_add_f32(&denom[d], ex);
}

// ---------- edge pass 3 (F=128): wave per edge, alpha-weighted scatter-add ----------
__global__ void agg128(const float* __restrict__ H, const float* __restrict__ ex,
                       const float* __restrict__ denom, const int* __restrict__ src,
                       const int* __restrict__ dst, int E, int nT,
                       float* __restrict__ agg) {
  int i = blockIdx.x * 8 + (threadIdx.x >> 5);
  if (i >= nT) return;
  int lane = threadIdx.x & 31;
  int s, d;
  if (i < E) { s = src[i]; d = dst[i]; } else { s = d = i - E; }
  float alpha = ex[i] / denom[d];
  float4 hv = ((const float4*)(H + (size_t)s * 128))[lane];
  float* o = agg + (size_t)d * 128 + lane * 4;
  atomic_add_f32(o + 0, hv.x * alpha);
  atomic_add_f32(o + 1, hv.y * alpha);
  atomic_add_f32(o + 2, hv.z * alpha);
  atomic_add_f32(o + 3, hv.w * alpha);
}

// ---------- bias + ReLU, in place on agg1 ----------
__global__ void bias_relu(float* __restrict__ a, const float* __restrict__ b, size_t n) {
  size_t i = (size_t)blockIdx.x * blockDim.x + threadIdx.x;
  if (i >= n) return;
  float v = a[i] + b[i & 127];
  a[i] = (v > 0.0f) ? v : 0.0f;
}

// ---------- layer 2: h2[N,2] = h1b @ W2[128,2]; plus e2 coefficients (wave per node) ----
__global__ void node_h2e2(const float* __restrict__ H, const float* __restrict__ W2,
                          const float* __restrict__ a2s, const float* __restrict__ a2d,
                          float* __restrict__ h2, float* __restrict__ e2s,
                          float* __restrict__ e2d, int N) {
  int n = blockIdx.x * 8 + (threadIdx.x >> 5);
  if (n >= N) return;
  int lane = threadIdx.x & 31;
  float4 hv = ((const float4*)(H + (size_t)n * 128))[lane];
  int c0 = lane * 4;
  float s0 = hv.x * W2[(c0 + 0) * 2 + 0] + hv.y * W2[(c0 + 1) * 2 + 0] +
             hv.z * W2[(c0 + 2) * 2 + 0] + hv.w * W2[(c0 + 3) * 2 + 0];
  float s1 = hv.x * W2[(c0 + 0) * 2 + 1] + hv.y * W2[(c0 + 1) * 2 + 1] +
             hv.z * W2[(c0 + 2) * 2 + 1] + hv.w * W2[(c0 + 3) * 2 + 1];
  s0 = warp_sum(s0);
  s1 = warp_sum(s1);
  if (lane == 0) {
    h2[2 * n + 0] = s0;
    h2[2 * n + 1] = s1;
    e2s[n] = s0 * a2s[0] + s1 * a2s[1];
    e2d[n] = s0 * a2d[0] + s1 * a2d[1];
  }
}

// ---------- edge pass 3 (F=2): thread per edge ----------
__global__ void agg2(const float* __restrict__ h2, const float* __restrict__ ex,
                     const float* __restrict__ denom, const int* __restrict__ src,
                     const int* __restrict__ dst, int E, int nT,
                     float* __restrict__ out) {
  int i = blockIdx.x * blockDim.x + threadIdx.x;
  if (i >= nT) return;
  int s, d;
  if (i < E) { s = src[i]; d = dst[i]; } else { s = d = i - E; }
  float alpha = ex[i] / denom[d];
  atomic_add_f32(&out[2 * d + 0], h2[2 * s + 0] * alpha);
  atomic_add_f32(&out[2 * d + 1], h2[2 * s + 1] * alpha);
}

__global__ void bias2(float* __restrict__ out, const float* __restrict__ b, int n2) {
  int i = blockIdx.x * blockDim.x + threadIdx.x;
  if (i >= n2) return;
  out[i] += b[i & 1];
}

// ---------- host ----------

extern "C" void kernel_launch(void* const* d_in, const int* in_sizes, int n_in,
                              void* d_out, int out_size, void* d_ws, size_t ws_size,
                              hipStream_t stream) {
  const float* x   = (const float*)d_in[0];
  const int*  eidx = (const int*)d_in[1];
  const float* W1  = (const float*)d_in[2];
  const float* a1s = (const float*)d_in[3];
  const float* a1d = (const float*)d_in[4];
  const float* b1  = (const float*)d_in[5];
  const float* W2  = (const float*)d_in[6];
  const float* a2s = (const float*)d_in[7];
  const float* a2d = (const float*)d_in[8];
  const float* b2  = (const float*)d_in[9];

  const int N  = in_sizes[0] / 256;
  const int E  = in_sizes[1] / 2;
  const int nT = E + N;
  const int* src = eidx;
  const int* dst = eidx + E;
  float* out = (float*)d_out;

  char* w = (char*)d_ws;
  // --- contiguous zero-initialized block: agg1 | mkey1 | den1 | mkey2 | den2 ---
  float*    zblk   = (float*)w;
  float*    ag1    = (float*)w;    w += (size_t)N * 128 * sizeof(float);
  unsigned* mkey1  = (unsigned*)w; w += (size_t)N * sizeof(unsigned);
  float*    den1   = (float*)w;    w += (size_t)N * sizeof(float);
  unsigned* mkey2  = (unsigned*)w; w += (size_t)N * sizeof(unsigned);
  float*    den2   = (float*)w;    w += (size_t)N * sizeof(float);
  const size_t zcount = (size_t)N * 132;   // elements in the zero block
  // --- rest of scratch ---
  float*    h1     = (float*)w;    w += (size_t)N * 128 * sizeof(float);
  float*    ebuf   = (float*)w;    w += (size_t)nT * sizeof(float);
  float*    e1s    = (float*)w;    w += (size_t)N * sizeof(float);
  float*    e1d    = (float*)w;    w += (size_t)N * sizeof(float);
  float*    h2     = (float*)w;    w += (size_t)N * 2 * sizeof(float);
  float*    e2s    = (float*)w;    w += (size_t)N * sizeof(float);
  float*    e2d    = (float*)w;    w += (size_t)N * sizeof(float);

  const int TB = 256;
  // re-initialize all accumulators every call (graph-replay safe); key 0 == -inf
  zero_f32<<<2048, TB, 0, stream>>>(zblk, zcount);
  zero_f32<<<256,  TB, 0, stream>>>(out, (size_t)N * 2);

  // layer 1
  gemm1_wmma<<<(N + 15) / 16, TB, 0, stream>>>(x, W1, h1, N);
  node_e1<<<(N + 7) / 8, TB, 0, stream>>>(h1, a1s, a1d, e1s, e1d, N);
  edge_logits<<<(nT + TB - 1) / TB, TB, 0, stream>>>(e1s, e1d, src, dst, E, nT, ebuf, mkey1);
  edge_exp<<<(nT + TB - 1) / TB, TB, 0, stream>>>(ebuf, mkey1, dst, E, nT, ebuf, den1);
  agg128<<<(nT + 7) / 8, TB, 0, stream>>>(h1, ebuf, den1, src, dst, E, nT, ag1);
  bias_relu<<<(int)(((size_t)N * 128 + TB - 1) / TB), TB, 0, stream>>>(ag1, b1, (size_t)N * 128);

  // layer 2
  node_h2e2<<<(N + 7) / 8, TB, 0, stream>>>(ag1, W2, a2s, a2d, h2, e2s, e2d, N);
  edge_logits<<<(nT + TB - 1) / TB, TB, 0, stream>>>(e2s, e2d, src, dst, E, nT, ebuf, mkey2);
  edge_exp<<<(nT + TB - 1) / TB, TB, 0, stream>>>(ebuf, mkey2, dst, E, nT, ebuf, den2);
  agg2<<<(nT + TB - 1) / TB, TB, 0, stream>>>(h2, ebuf, den2, src, dst, E, nT, out);
  bias2<<<(2 * N + TB - 1) / TB, TB, 0, stream>>>(out, b2, 2 * N);
}